// GaussianShape_21449066676417
// MI455X (gfx1250) — compile-verified
//
#include <hip/hip_runtime.h>
#include <hip/hip_bf16.h>

typedef __attribute__((ext_vector_type(2))) float v2f;
typedef __attribute__((ext_vector_type(8))) float v8f;

#define NFILT 128
#define FILTER_SIZE 1025
#define FBINS 1025            // F = NFFT/2 + 1
#define HALFW 2               // HALF = (1025//128)//4
#define BATCH 32
#define TLEN 3000
#define KMAX 256              // max compacted K span per 16-filter tile
#define MTILES 8              // NFILT / 16
#define CN_LO (30.0f / 80000.0f)

// ---------------------------------------------------------------------------
// Phase 1: filter-bank construction (band-compacted), headers, and loss.
// One block of 128 threads; negligible runtime.
// ---------------------------------------------------------------------------
__global__ __launch_bounds__(128) void prep_kernel(
    const float* __restrict__ w,     // (128, 2): [:,0]=cn, [:,1]=sigma
    float* __restrict__ Wc,          // [MTILES][KMAX][16] compacted filters
    int* __restrict__ hdr,           // [MTILES][2]: kStart, kSteps(=span/4)
    float* __restrict__ loss_out)
{
    __shared__ float s_c[NFILT];     // cumsum of clipped cn
    __shared__ float s_sig[NFILT];   // clipped sigma
    __shared__ int   s_bl[NFILT];
    __shared__ int   s_bh[NFILT];
    __shared__ int   s_ks[MTILES];

    const int tid = threadIdx.x;

    const float cnr = w[2 * tid];
    const float sgr = w[2 * tid + 1];
    const float cnc = fminf(fmaxf(cnr, CN_LO), 1.0f);
    const float sgc = fminf(fmaxf(sgr, 1e-4f), 0.05f);
    s_c[tid]   = cnc;
    s_sig[tid] = sgc;
    __syncthreads();

    if (tid == 0) {                  // serial cumsum over 128 elems: trivial
        float run = 0.0f;
        for (int i = 0; i < NFILT; ++i) { run += s_c[i]; s_c[i] = run; }
    }
    __syncthreads();

    // Per-filter band (matches reference exactly)
    {
        const float c  = s_c[tid];
        const int   ic = (int)floorf(c * (float)FILTER_SIZE);
        const int   lo = ic - HALFW;
        const int   bl = (lo >= 0 && lo < FILTER_SIZE) ? lo : 0;
        const int   hi = ic + HALFW;
        const int   bh = (hi < FILTER_SIZE) ? hi + 1 : FILTER_SIZE - 1;
        s_bl[tid] = bl;
        s_bh[tid] = bh;
    }
    __syncthreads();

    // Per-tile compacted K window
    if (tid < MTILES) {
        int mn = 0x7fffffff, mx = 0;
        for (int i = 0; i < 16; ++i) {
            const int n = tid * 16 + i;
            mn = min(mn, s_bl[n]);
            mx = max(mx, s_bh[n]);
        }
        int ks   = mn & ~3;                     // align to K=4
        int span = mx - ks;
        span = (span + 3) & ~3;
        if (span < 4)    span = 4;
        if (span > KMAX) span = KMAX;
        hdr[2 * tid]     = ks;
        hdr[2 * tid + 1] = span >> 2;
        s_ks[tid] = ks;
    }

    // Loss: uses UNCLIPPED cn / sigma (matches reference)
    if (tid == 0) {
        float ssum = 0.0f, pl = 0.0f;
        for (int i = 0; i < NFILT; ++i) {
            const float a = w[2 * i];
            const float b = w[2 * i + 1];
            ssum += a;
            pl += (a < 0.0f ? -a : 0.0f) + (b < 0.0f ? -b : 0.0f);
        }
        float dc;
        if (ssum >= 0.2f && ssum <= 0.5f) dc = 0.0f;
        else dc = (ssum < 0.2f ? expf(-ssum) : expf(ssum)) * 100.0f;
        loss_out[0] = pl * 1000.0f + dc;
    }
    __syncthreads();

    // Materialize compacted filter blocks: Wc[tile][k][nl].
    // Rows with j >= band (including j >= FBINS) are zero -> main kernel may
    // feed garbage B values there (clamped addresses), product is exactly 0.
    for (int idx = tid; idx < MTILES * KMAX * 16; idx += 128) {
        const int tile = idx / (KMAX * 16);
        const int rem  = idx - tile * (KMAX * 16);
        const int k    = rem >> 4;
        const int nl   = rem & 15;
        const int n    = tile * 16 + nl;
        const int j    = s_ks[tile] + k;
        float val = 0.0f;
        if (j >= s_bl[n] && j < s_bh[n]) {      // bh <= 1025 so j < FBINS implied
            const float t = (float)j / (float)FILTER_SIZE;
            const float z = (t - s_c[n]) / s_sig[n];
            val = expf(-0.5f * z * z);
        }
        Wc[idx] = val;
    }
}

// ---------------------------------------------------------------------------
// Phase 2: banded GEMM via fp32 WMMA.
// grid = (ceil(T/64), MTILES, BATCH); block = 128 threads = 4 waves.
// Each wave computes one 16(filters) x 16(time) tile; K loop over the tile's
// compacted band in steps of 4 with v_wmma_f32_16x16x4_f32.
// Hot loop is branch-free: bounds handled by address clamping (A-side zeros
// make the clamped products exact), so EXEC is all-ones throughout.
// ---------------------------------------------------------------------------
__global__ __launch_bounds__(128) void band_wmma_kernel(
    const float* __restrict__ X,     // (B, F, T) fp32
    const float* __restrict__ Wc,
    const int* __restrict__ hdr,
    float* __restrict__ out)         // (B, NFILT, T) fp32
{
    __shared__ float sW[KMAX * 16];  // 16 KB filter tile

    const int tile = blockIdx.y;
    const int b    = blockIdx.z;
    const int ks   = hdr[2 * tile];
    const int kn4  = hdr[2 * tile + 1];
    const int span = kn4 << 2;

    // Stage this tile's compacted filters into LDS (shared by 4 waves)
    const float* Wt = Wc + tile * (KMAX * 16);
    for (int i = threadIdx.x; i < span * 16; i += 128) sW[i] = Wt[i];
    __syncthreads();

    const int wave = threadIdx.x >> 5;
    const int lane = threadIdx.x & 31;
    const int half = lane >> 4;        // 0: lanes 0-15, 1: lanes 16-31
    const int lm   = lane & 15;
    const int col  = blockIdx.x * 64 + wave * 16 + lm;
    const bool colOk = (col < TLEN);
    const int colC = colOk ? col : (TLEN - 1);   // clamped load column

    const float* Xb = X + (size_t)b * (size_t)FBINS * TLEN;

    // Iterations where ks + 4*kk + 3 <= FBINS-1 need no row clamp at all.
    int kkSafe = (FBINS - 4 - ks) >= 0 ? (((FBINS - 4 - ks) >> 2) + 1) : 0;
    if (kkSafe > kn4) kkSafe = kn4;

    v8f acc = {};

    // Per-lane streaming pointer: row j0 = ks + 2*half, column colC.
    const float* p = Xb + (size_t)(ks + (half << 1)) * TLEN + colC;
    int lidx = (half << 1) * 16 + lm;            // LDS index for A fragment

    int kk = 0;
    for (; kk < kkSafe; ++kk) {
        v2f a;
        a.x = sW[lidx];
        a.y = sW[lidx + 16];
        v2f bb;
        bb.x = __builtin_nontemporal_load(p);          // row j0
        bb.y = __builtin_nontemporal_load(p + TLEN);   // row j0+1
        acc = __builtin_amdgcn_wmma_f32_16x16x4_f32(
            false, a, false, bb, (short)0, acc, false, false);
        p += 4 * TLEN;
        lidx += 64;
    }

    // Tail (<= 2 iterations): clamp row indices into [0, FBINS-1]; the
    // corresponding A entries are zero, so clamped loads are exact.
    for (; kk < kn4; ++kk) {
        const int ja = ks + (kk << 2) + (half << 1);
        const int j1 = min(ja,     FBINS - 1);
        const int j2 = min(ja + 1, FBINS - 1);
        v2f a;
        a.x = sW[((ja - ks)    ) * 16 + lm];
        a.y = sW[((ja - ks) + 1) * 16 + lm];
        v2f bb;
        bb.x = __builtin_nontemporal_load(Xb + (size_t)j1 * TLEN + colC);
        bb.y = __builtin_nontemporal_load(Xb + (size_t)j2 * TLEN + colC);
        acc = __builtin_amdgcn_wmma_f32_16x16x4_f32(
            false, a, false, bb, (short)0, acc, false, false);
    }

    // D (16x16 f32): VGPR r -> row M = r + 8*half, col N = lane%16
    if (colOk) {
        float* Ob = out + ((size_t)b * NFILT + tile * 16) * TLEN + col;
        #pragma unroll
        for (int r = 0; r < 8; ++r) {
            __builtin_nontemporal_store(acc[r], Ob + (size_t)(r + half * 8) * TLEN);
        }
    }
}

// ---------------------------------------------------------------------------
extern "C" void kernel_launch(void* const* d_in, const int* in_sizes, int n_in,
                              void* d_out, int out_size, void* d_ws, size_t ws_size,
                              hipStream_t stream) {
    const float* X = (const float*)d_in[0];   // inputs (32, 1025, 3000) fp32
    const float* w = (const float*)d_in[1];   // w (128, 2) fp32

    float* out  = (float*)d_out;              // res (32,128,3000) then loss
    float* Wc   = (float*)d_ws;               // 128 KB compacted filters
    int*   hdr  = (int*)((char*)d_ws + (size_t)MTILES * KMAX * 16 * sizeof(float));
    float* loss = out + (size_t)BATCH * NFILT * TLEN;

    prep_kernel<<<1, 128, 0, stream>>>(w, Wc, hdr, loss);

    dim3 grid((TLEN + 63) / 64, MTILES, BATCH);
    band_wmma_kernel<<<grid, 128, 0, stream>>>(X, Wc, hdr, out);
}